// Encoder_35682588295653
// MI455X (gfx1250) — compile-verified
//
#include <hip/hip_runtime.h>
#include <math.h>

#define BB 64
#define TT 48
#define NN 184
#define FF 16
#define HID 64
#define EMB 16
#define DEG 12
#define BN (BB*NN)            // 11776
#define EE (BN*DEG)           // 141312
#define SROW 72               // [xf 33 | pad 3 | agg 33 | pad 3]
#define HSIZE ((size_t)BB*TT*NN*HID)

typedef __attribute__((ext_vector_type(2))) float v2f;
typedef __attribute__((ext_vector_type(8))) float v8f;

static __device__ __forceinline__ v8f wmma4(v2f a, v2f b, v8f c) {
  // D = A(16x4,f32) * B(4x16,f32) + C(16x16,f32)  — exact fp32 matrix op
  return __builtin_amdgcn_wmma_f32_16x16x4_f32(false, a, false, b, (short)0, c,
                                               false, false);
}
static __device__ __forceinline__ float sigmoidf_(float x) {
  return 1.0f / (1.0f + __expf(-x));
}

// ---------------------------------------------------------------------------
// One-time per-launch prep: padded weights + zero hidden state.
// Wc   (72 x 64)  : rows 0-32 = W_root, 33-35 = 0, 36-68 = W_rel, 69-71 = 0
// Wihp (100 x 192): rows 0-32 = W_ih[0:33], 33-35 = 0, 36-99 = W_ih[33:97]
// ---------------------------------------------------------------------------
__global__ void prep_kernel(const float* __restrict__ W_root,
                            const float* __restrict__ W_rel,
                            const float* __restrict__ W_ih,
                            float* __restrict__ Wc, float* __restrict__ Wihp,
                            float* __restrict__ hbuf) {
  int tid = blockIdx.x * blockDim.x + threadIdx.x;
  int stride = gridDim.x * blockDim.x;
  for (int i = tid; i < 72 * 64; i += stride) {
    int r = i / 64, c = i % 64;
    float v = 0.f;
    if (r < 33) v = W_root[r * 64 + c];
    else if (r >= 36 && r < 69) v = W_rel[(r - 36) * 64 + c];
    Wc[i] = v;
  }
  for (int i = tid; i < 100 * 192; i += stride) {
    int r = i / 192, c = i % 192;
    float v = 0.f;
    if (r < 33) v = W_ih[r * 192 + c];
    else if (r >= 36) v = W_ih[(r - 3) * 192 + c];
    Wihp[i] = v;
  }
  for (int i = tid; i < BN * HID; i += stride) hbuf[i] = 0.f;
}

// ---------------------------------------------------------------------------
// Per-step input assembly: rowbuf[row] = [xn, y, Xc(15), emb(16), 0(3), agg=0(36)]
// ---------------------------------------------------------------------------
__global__ void step_assemble(const float* __restrict__ X,
                              const float* __restrict__ y,
                              const float* __restrict__ emb,
                              const float* __restrict__ outp,
                              float* __restrict__ rowbuf, int t) {
  int row = blockIdx.x * blockDim.x + threadIdx.x;
  if (row >= BN) return;
  int bb = row / NN, nn = row % NN;
  size_t xbase = ((size_t)(bb * TT + t) * NN + nn) * FF;
  float* rp = rowbuf + (size_t)row * SROW;
  rp[0] = (t == 0) ? 0.f : outp[HSIZE + (size_t)(bb * TT + (t - 1)) * NN + nn];
  rp[1] = y[(size_t)(bb * TT + t) * NN + nn];
#pragma unroll
  for (int f = 0; f < 15; ++f) rp[2 + f] = X[xbase + f];
  int idx = (int)X[xbase + 15];
#pragma unroll
  for (int e = 0; e < 16; ++e) rp[17 + e] = emb[idx * EMB + e];
#pragma unroll
  for (int k = 33; k < 72; ++k) rp[k] = 0.f;
}

// ---------------------------------------------------------------------------
// Graph aggregation: agg[dst] += xf[src] * w   (one thread per edge*feature)
// ---------------------------------------------------------------------------
__global__ void step_edges(const int* __restrict__ ei,
                           const float* __restrict__ ew,
                           float* __restrict__ rowbuf) {
  int tid = blockIdx.x * blockDim.x + threadIdx.x;
  if (tid >= EE * 33) return;
  int e = tid / 33;
  int f = tid - e * 33;
  int s = ei[e];
  int d = ei[EE + e];
  atomicAdd(&rowbuf[(size_t)d * SROW + 36 + f],
            rowbuf[(size_t)s * SROW + f] * ew[e]);
}

// ---------------------------------------------------------------------------
// Fused per-step compute: GEMM1 (WMMA f32) -> g ; GEMM2 (WMMA f32) -> GRU ->
// h, H[t], preds[t].  256 threads = 8 waves handle 32 rows (2 M-tiles).
// ---------------------------------------------------------------------------
__global__ __launch_bounds__(256) void step_compute(
    const float* __restrict__ rowbuf, const float* __restrict__ Wc,
    const float* __restrict__ Wihp, const float* __restrict__ W_hh,
    const float* __restrict__ b_conv, const float* __restrict__ b_ih,
    const float* __restrict__ b_hh, const float* __restrict__ fc_w,
    const float* __restrict__ fc_b, float* __restrict__ hbuf,
    float* __restrict__ outp, int t) {
  __shared__ float gbuf[32][65];   // g (then reused for h in pred-reduce)
  __shared__ float hold[32][65];   // previous hidden state
  __shared__ float rzb[32][130];   // r | z gates
  __shared__ float xgb[32][65];    // xg
  __shared__ float hgb[32][65];    // hg

  const int tid = threadIdx.x;
  const int wave = tid >> 5;
  const int lane = tid & 31;
  const int rowblk = blockIdx.x * 32;

  // load h_old into LDS
  for (int i = tid; i < 32 * 64; i += 256) {
    int r = i >> 6, c = i & 63;
    hold[r][c] = hbuf[(size_t)(rowblk + r) * HID + c];
  }
  __syncthreads();

  const int r16 = lane & 15;
  const int koff = (lane >> 4) << 1;  // 0 (lanes 0-15) or 2 (lanes 16-31)
  const int rbase = ((lane >> 4) << 3);

  // ---- Stage 1: g = sigmoid([xf|agg] @ Wc + b_conv), K = 72 ----
  {
    const int mt = wave >> 2;  // 0..1 -> which 16-row tile
    const int nt = wave & 3;   // 0..3 -> 16-col tile of 64
    const int arow = rowblk + mt * 16 + r16;
    const int col = nt * 16 + r16;
    v8f acc = {};
#pragma unroll
    for (int kc = 0; kc < 18; ++kc) {
      int kb = kc * 4 + koff;
      v2f a, b;
      a.x = rowbuf[(size_t)arow * SROW + kb];
      a.y = rowbuf[(size_t)arow * SROW + kb + 1];
      b.x = Wc[kb * 64 + col];
      b.y = Wc[(kb + 1) * 64 + col];
      acc = wmma4(a, b, acc);
    }
    float bc = b_conv[col];
#pragma unroll
    for (int v = 0; v < 8; ++v)
      gbuf[mt * 16 + rbase + v][col] = sigmoidf_(acc[v] + bc);
  }
  __syncthreads();

  // ---- Stage 2: gx = [xf|g] @ W_ih, gh = h @ W_hh (24 tiles, 3 per wave) ----
  for (int task = wave; task < 24; task += 8) {
    const int mt = task / 12;
    const int nt = task % 12;
    const int arow = rowblk + mt * 16 + r16;
    const int lrow = mt * 16 + r16;
    const int col = nt * 16 + r16;
    v8f ax = {};
    v8f ah = {};
#pragma unroll
    for (int kc = 0; kc < 9; ++kc) {  // xf part, K=36 (zero-padded)
      int kb = kc * 4 + koff;
      v2f a, b;
      a.x = rowbuf[(size_t)arow * SROW + kb];
      a.y = rowbuf[(size_t)arow * SROW + kb + 1];
      b.x = Wihp[kb * 192 + col];
      b.y = Wihp[(kb + 1) * 192 + col];
      ax = wmma4(a, b, ax);
    }
#pragma unroll
    for (int kc = 0; kc < 16; ++kc) {  // g part (K=64) + h part (K=64)
      int kb = kc * 4 + koff;
      v2f a, b;
      a.x = gbuf[lrow][kb];
      a.y = gbuf[lrow][kb + 1];
      b.x = Wihp[(36 + kb) * 192 + col];
      b.y = Wihp[(37 + kb) * 192 + col];
      ax = wmma4(a, b, ax);
      a.x = hold[lrow][kb];
      a.y = hold[lrow][kb + 1];
      b.x = W_hh[kb * 192 + col];
      b.y = W_hh[(kb + 1) * 192 + col];
      ah = wmma4(a, b, ah);
    }
    float bi = b_ih[col], bh = b_hh[col];
    if (nt < 8) {  // r and z gate columns: sum then sigmoid
#pragma unroll
      for (int v = 0; v < 8; ++v)
        rzb[mt * 16 + rbase + v][col] = sigmoidf_((ax[v] + bi) + (ah[v] + bh));
    } else {  // candidate columns: keep xg, hg separate
#pragma unroll
      for (int v = 0; v < 8; ++v) {
        xgb[mt * 16 + rbase + v][col - 128] = ax[v] + bi;
        hgb[mt * 16 + rbase + v][col - 128] = ah[v] + bh;
      }
    }
  }
  __syncthreads();

  // ---- Stage 3: GRU pointwise + outputs ----
  {
    const int r = tid >> 3;          // row 0..31
    const int cb = (tid & 7) * 8;    // 8 cols per thread
    const int grow = rowblk + r;
    const int bb = grow / NN, nn = grow % NN;
    float* Hout = outp + ((size_t)(bb * TT + t) * NN + nn) * HID;
#pragma unroll
    for (int j = 0; j < 8; ++j) {
      int c = cb + j;
      float rr = rzb[r][c];
      float zz = rzb[r][64 + c];
      float nv = tanhf(xgb[r][c] + rr * hgb[r][c]);
      float h = (1.0f - zz) * nv + zz * hold[r][c];
      hbuf[(size_t)grow * HID + c] = h;
      Hout[c] = h;
      gbuf[r][c] = h;  // stash for the fc reduction
    }
  }
  __syncthreads();
  if (tid < 32) {  // per-row prediction: dot(h, fc_w) + fc_b
    const int grow = rowblk + tid;
    const int bb = grow / NN, nn = grow % NN;
    float s = 0.f;
#pragma unroll
    for (int c = 0; c < 64; ++c) s += gbuf[tid][c] * fc_w[c];
    outp[HSIZE + (size_t)(bb * TT + t) * NN + nn] = s + fc_b[0];
  }
}

// ---------------------------------------------------------------------------
extern "C" void kernel_launch(void* const* d_in, const int* in_sizes, int n_in,
                              void* d_out, int out_size, void* d_ws,
                              size_t ws_size, hipStream_t stream) {
  const float* X = (const float*)d_in[0];
  const float* y = (const float*)d_in[1];
  const float* emb = (const float*)d_in[2];
  const float* W_root = (const float*)d_in[3];
  const float* W_rel = (const float*)d_in[4];
  const float* b_conv = (const float*)d_in[5];
  const float* W_ih = (const float*)d_in[6];
  const float* W_hh = (const float*)d_in[7];
  const float* b_ih = (const float*)d_in[8];
  const float* b_hh = (const float*)d_in[9];
  const float* fc_w = (const float*)d_in[10];
  const float* fc_b = (const float*)d_in[11];
  const float* ew = (const float*)d_in[12];
  const int* ei = (const int*)d_in[13];
  float* outp = (float*)d_out;

  float* ws = (float*)d_ws;
  float* rowbuf = ws;                            // BN*72   floats
  float* hbuf = rowbuf + (size_t)BN * SROW;      // BN*64   floats
  float* Wc = hbuf + (size_t)BN * HID;           // 72*64   floats
  float* Wihp = Wc + 72 * 64;                    // 100*192 floats  (~6.5 MB total)

  prep_kernel<<<512, 256, 0, stream>>>(W_root, W_rel, W_ih, Wc, Wihp, hbuf);
  for (int t = 0; t < TT; ++t) {
    step_assemble<<<BN / 256, 256, 0, stream>>>(X, y, emb, outp, rowbuf, t);
    step_edges<<<(EE * 33 + 255) / 256, 256, 0, stream>>>(ei, ew, rowbuf);
    step_compute<<<BN / 32, 256, 0, stream>>>(rowbuf, Wc, Wihp, W_hh, b_conv,
                                              b_ih, b_hh, fc_w, fc_b, hbuf,
                                              outp, t);
  }
}